// TripletLoss_59390807769654
// MI455X (gfx1250) — compile-verified
//
#include <hip/hip_runtime.h>
#include <hip/hip_bf16.h>

// Triplet margin loss (batch-all), B=512 anchors, D=512 dims, fp32.
// Pipeline: row sq-norms -> WMMA fp32 Gram + distance epilogue -> per-anchor
// triplet relu-sum -> deterministic final reduce.

#define BATCH   512
#define DIM     512
#define MARGIN  0.3f

typedef float v2f __attribute__((ext_vector_type(2)));
typedef float v8f __attribute__((ext_vector_type(8)));

// ---------------------------------------------------------------------------
// Kernel 1: sq[i] = ||E[i,:]||^2.  One block per row, 128 threads.
// ---------------------------------------------------------------------------
__global__ void tl_sq_kernel(const float* __restrict__ E, float* __restrict__ sq) {
    __shared__ float red[128];
    const int row = blockIdx.x;
    const int tid = threadIdx.x;
    float s = 0.0f;
    const float* r = E + (size_t)row * DIM;
    for (int t = tid; t < DIM; t += 128) {
        float v = r[t];
        s += v * v;
    }
    red[tid] = s;
    __syncthreads();
    for (int off = 64; off > 0; off >>= 1) {
        if (tid < off) red[tid] += red[tid + off];
        __syncthreads();
    }
    if (tid == 0) sq[row] = red[0];
}

// ---------------------------------------------------------------------------
// Kernel 2: Gram via V_WMMA_F32_16X16X4_F32, distance epilogue.
// One wave32 per 16x16 output tile; 8 waves/block; 1024 tiles -> 128 blocks.
//
// Per-lane fragment addressing (ISA 7.12.2, 32-bit layouts), m=lane&15, h=lane>>4:
//   A (16x4, MxK): v0 = E[row_base+m][k+2h],   v1 = E[row_base+m][k+2h+1]
//   B (4x16, KxN): v0 = E[col_base+m][k+2h],   v1 = E[col_base+m][k+2h+1]
//   (identical formulas because both operands are row-blocks of E)
//   C (16x16):     c[v] = G[row_base+v+8h][col_base+m]
// ---------------------------------------------------------------------------
__global__ void tl_gram_dist_kernel(const float* __restrict__ E,
                                    const float* __restrict__ sq,
                                    float* __restrict__ dist) {
    const int wid  = threadIdx.x >> 5;
    const int lane = threadIdx.x & 31;
    const int tile = blockIdx.x * 8 + wid;          // 0..1023
    const int trow = tile >> 5;                     // 32 tile-cols per tile-row
    const int tcol = tile & 31;
    const int m = lane & 15;
    const int h = lane >> 4;

    const int row_base = trow * 16;
    const int col_base = tcol * 16;

    const float* arow = E + (size_t)(row_base + m) * DIM + 2 * h;
    const float* brow = E + (size_t)(col_base + m) * DIM + 2 * h;

    v8f c = {};
    #pragma unroll 4
    for (int k = 0; k < DIM; k += 4) {
        v2f a = *(const v2f*)(arow + k);            // 8B-aligned b64 load
        v2f b = *(const v2f*)(brow + k);
        // 8 args: (neg_a, A, neg_b, B, c_mod, C, reuse_a, reuse_b)
        c = __builtin_amdgcn_wmma_f32_16x16x4_f32(
                false, a, false, b, (short)0, c, false, false);
    }

    const int col = col_base + m;
    const float sqc = sq[col];
    #pragma unroll
    for (int v = 0; v < 8; ++v) {
        const int row = row_base + v + 8 * h;
        float d2 = sq[row] + sqc - 2.0f * c[v];
        d2 = fmaxf(d2, 0.0f);
        // Reference zero-mask semantics: exact-zero d2 -> distance 0.
        const float d = (d2 == 0.0f) ? 0.0f : sqrtf(d2);
        dist[(size_t)row * BATCH + col] = d;
    }
}

// ---------------------------------------------------------------------------
// Kernel 3: per-anchor batch-all triplet partial sums.
// One block (256 threads) per anchor; distance row + labels in LDS.
// partial_num[a] = sum_{p in P_a, n in N_a} relu(d[a,p] - d[a,n] + MARGIN)
// partial_cnt[a] = |P_a| * |N_a|
// ---------------------------------------------------------------------------
__global__ void tl_triplet_kernel(const float* __restrict__ dist,
                                  const int* __restrict__ labels,
                                  float* __restrict__ partial_num,
                                  unsigned int* __restrict__ partial_cnt) {
    __shared__ float sd[BATCH];
    __shared__ int   sl[BATCH];
    __shared__ float rs[256];
    __shared__ int   rp[256];
    __shared__ int   rn[256];

    const int a   = blockIdx.x;
    const int tid = threadIdx.x;

    for (int j = tid; j < BATCH; j += 256) {
        sd[j] = dist[(size_t)a * BATCH + j];
        sl[j] = labels[j];
    }
    __syncthreads();

    const int la = sl[a];
    float sum = 0.0f;
    int npos = 0, nneg = 0;

    for (int p = tid; p < BATCH; p += 256) {
        const int lp = sl[p];
        if (lp != la) { nneg++; continue; }
        if (p == a)   continue;
        npos++;
        const float dp = sd[p] + MARGIN;
        float local = 0.0f;
        for (int n = 0; n < BATCH; ++n) {
            if (sl[n] != la) local += fmaxf(dp - sd[n], 0.0f);
        }
        sum += local;
    }

    rs[tid] = sum; rp[tid] = npos; rn[tid] = nneg;
    __syncthreads();
    for (int off = 128; off > 0; off >>= 1) {
        if (tid < off) {
            rs[tid] += rs[tid + off];
            rp[tid] += rp[tid + off];
            rn[tid] += rn[tid + off];
        }
        __syncthreads();
    }
    if (tid == 0) {
        partial_num[a] = rs[0];
        partial_cnt[a] = (unsigned int)(rp[0] * rn[0]);
    }
}

// ---------------------------------------------------------------------------
// Kernel 4: deterministic final reduce over 512 anchor partials.
// ---------------------------------------------------------------------------
__global__ void tl_finalize_kernel(const float* __restrict__ partial_num,
                                   const unsigned int* __restrict__ partial_cnt,
                                   float* __restrict__ out) {
    __shared__ float rs[256];
    __shared__ unsigned int rc[256];
    const int tid = threadIdx.x;
    float s = 0.0f;
    unsigned int c = 0u;
    for (int j = tid; j < BATCH; j += 256) {
        s += partial_num[j];
        c += partial_cnt[j];
    }
    rs[tid] = s; rc[tid] = c;
    __syncthreads();
    for (int off = 128; off > 0; off >>= 1) {
        if (tid < off) { rs[tid] += rs[tid + off]; rc[tid] += rc[tid + off]; }
        __syncthreads();
    }
    if (tid == 0) {
        const unsigned int cnt = rc[0];
        out[0] = (cnt > 0u) ? (rs[0] / (float)cnt) : 0.0f;
    }
}

// ---------------------------------------------------------------------------
extern "C" void kernel_launch(void* const* d_in, const int* in_sizes, int n_in,
                              void* d_out, int out_size, void* d_ws, size_t ws_size,
                              hipStream_t stream) {
    (void)in_sizes; (void)n_in; (void)out_size; (void)ws_size;

    const float* E      = (const float*)d_in[0];   // [512,512] fp32
    const int*   labels = (const int*)d_in[1];     // [512] int
    float*       out    = (float*)d_out;           // scalar fp32

    char* ws = (char*)d_ws;
    float*        dist = (float*)ws;                                   // 1 MB
    float*        sq   = (float*)(ws + (size_t)BATCH * BATCH * 4);     // 2 KB
    float*        pnum = sq + BATCH;                                   // 2 KB
    unsigned int* pcnt = (unsigned int*)(pnum + BATCH);                // 2 KB

    tl_sq_kernel<<<BATCH, 128, 0, stream>>>(E, sq);

    const int tiles  = (BATCH / 16) * (BATCH / 16);   // 1024
    tl_gram_dist_kernel<<<tiles / 8, 256, 0, stream>>>(E, sq, dist);

    tl_triplet_kernel<<<BATCH, 256, 0, stream>>>(dist, labels, pnum, pcnt);

    tl_finalize_kernel<<<1, 256, 0, stream>>>(pnum, pcnt, out);
}